// ConvCapsule_10462540333706
// MI455X (gfx1250) — compile-verified
//
#include <hip/hip_runtime.h>
#include <math.h>

typedef __attribute__((ext_vector_type(2))) float v2f;
typedef __attribute__((ext_vector_type(8))) float v8f;

#define P_TOT   1152   // 32*6*6 input capsule positions
#define O_CAPS  10
#define E_DIM   16
#define D_DIM   8
#define BATCH   256
#define ITERS   3

// ---------------------------------------------------------------------------
// Kernel 1: u_hat[b,p,o,e] = sum_d x[b,p,d] * W[p,o,d,e]
// One wave per (p, btile): C tile = X[btile*16..+15, 0..7] x W[p,o] (8x16),
// via two V_WMMA_F32_16X16X4_F32 (K=0..3, K=4..7). A fragment reused over o.
// ---------------------------------------------------------------------------
__global__ __launch_bounds__(256) void uhat_wmma_kernel(
    const float* __restrict__ x, const float* __restrict__ W,
    float* __restrict__ uhat)
{
    const int tid  = threadIdx.x;
    const int lane = tid & 31;
    const int wid  = blockIdx.x * 8 + (tid >> 5);   // 0 .. 18431
    const int p     = wid >> 4;                     // 0 .. 1151
    const int btile = wid & 15;                     // 0 .. 15
    const int m   = lane & 15;                      // row (A) / col (B,C)
    const int klo = lane >> 4;                      // K half select

    // A fragment: rows b = btile*16 + m, K = d.
    // VGPR0 holds K = klo*2, VGPR1 holds K = klo*2+1  (ISA 16x4 f32 layout)
    const int brow = btile * 16 + m;
    const float* xb = x + ((size_t)brow * P_TOT + p) * D_DIM;
    v2f a0 = *(const v2f*)(xb + klo * 2);        // K = klo*2,   klo*2+1
    v2f a1 = *(const v2f*)(xb + 4 + klo * 2);    // K = 4+klo*2, 5+klo*2

    const float* wp = W + (size_t)p * O_CAPS * D_DIM * E_DIM;

    for (int o = 0; o < O_CAPS; ++o) {
        const float* wpo = wp + o * D_DIM * E_DIM;
        // B fragment (4x16, symmetric layout): VGPR v = row K=klo*2+v, col N=m
        v2f b0, b1;
        b0.x = wpo[(klo * 2 + 0) * E_DIM + m];
        b0.y = wpo[(klo * 2 + 1) * E_DIM + m];
        b1.x = wpo[(4 + klo * 2 + 0) * E_DIM + m];
        b1.y = wpo[(4 + klo * 2 + 1) * E_DIM + m];

        v8f c = {0.f, 0.f, 0.f, 0.f, 0.f, 0.f, 0.f, 0.f};
        c = __builtin_amdgcn_wmma_f32_16x16x4_f32(false, a0, false, b0,
                                                  (short)0, c, false, false);
        c = __builtin_amdgcn_wmma_f32_16x16x4_f32(false, a1, false, b1,
                                                  (short)0, c, false, false);

        // C layout: VGPR r -> row M = r + klo*8, col N = m
        #pragma unroll
        for (int r = 0; r < 8; ++r) {
            const int bout = btile * 16 + r + klo * 8;
            uhat[(((size_t)bout * P_TOT + p) * O_CAPS + o) * E_DIM + m] = c[r];
        }
    }
}

// ---------------------------------------------------------------------------
// Kernel 2: dynamic routing. One block per batch element.
// 384 threads: each owns 3 positions p (logits in registers),
// softmax coefficients cc[P_TOT*O_CAPS] in LDS (46 KB).
// ---------------------------------------------------------------------------
__global__ __launch_bounds__(384) void routing_kernel(
    const float* __restrict__ uhat, float* __restrict__ out)
{
    __shared__ float cc[P_TOT * O_CAPS];      // 46080 B
    __shared__ float sbuf[O_CAPS * E_DIM];
    __shared__ float vbuf[O_CAPS * E_DIM];
    __shared__ float scl[O_CAPS];

    const int b   = blockIdx.x;
    const int tid = threadIdx.x;
    const float* U = uhat + (size_t)b * P_TOT * O_CAPS * E_DIM;

    float lg[3][O_CAPS];
    #pragma unroll
    for (int pi = 0; pi < 3; ++pi)
        #pragma unroll
        for (int o = 0; o < O_CAPS; ++o) lg[pi][o] = 0.f;

    for (int it = 0; it < ITERS; ++it) {
        // c = softmax(logits) over O, per owned position p
        #pragma unroll
        for (int pi = 0; pi < 3; ++pi) {
            const int p = tid + pi * 384;
            float mx = lg[pi][0];
            #pragma unroll
            for (int o = 1; o < O_CAPS; ++o) mx = fmaxf(mx, lg[pi][o]);
            float ex[O_CAPS];
            float sum = 0.f;
            #pragma unroll
            for (int o = 0; o < O_CAPS; ++o) {
                ex[o] = expf(lg[pi][o] - mx);
                sum += ex[o];
            }
            const float inv = 1.f / sum;
            #pragma unroll
            for (int o = 0; o < O_CAPS; ++o) cc[p * O_CAPS + o] = ex[o] * inv;
        }
        __syncthreads();

        // s[o,e] = sum_p c[p,o] * u_hat[p,o,e]
        if (tid < O_CAPS * E_DIM) {
            const int o = tid >> 4, e = tid & 15;
            float acc = 0.f;
            for (int p = 0; p < P_TOT; ++p)
                acc += cc[p * O_CAPS + o] * U[((size_t)p * O_CAPS + o) * E_DIM + e];
            sbuf[tid] = acc;
        }
        __syncthreads();

        // squash scale per output capsule
        if (tid < O_CAPS) {
            float sn = 0.f;
            #pragma unroll
            for (int e = 0; e < E_DIM; ++e) {
                const float t = sbuf[tid * E_DIM + e];
                sn += t * t;
            }
            scl[tid] = sn / (1.f + sn) / (sqrtf(sn) + 1e-6f);
        }
        __syncthreads();
        if (tid < O_CAPS * E_DIM) vbuf[tid] = sbuf[tid] * scl[tid >> 4];
        __syncthreads();

        // logit update: b[p,o] += <u_hat[p,o,:], v[o,:]>
        if (it < ITERS - 1) {
            #pragma unroll
            for (int pi = 0; pi < 3; ++pi) {
                const int p = tid + pi * 384;
                #pragma unroll
                for (int o = 0; o < O_CAPS; ++o) {
                    const float* up = U + ((size_t)p * O_CAPS + o) * E_DIM;
                    float d = 0.f;
                    #pragma unroll
                    for (int e = 0; e < E_DIM; ++e) d += up[e] * vbuf[o * E_DIM + e];
                    lg[pi][o] += d;
                }
            }
            __syncthreads();
        }
    }

    if (tid < O_CAPS * E_DIM)
        out[(size_t)b * O_CAPS * E_DIM + tid] = vbuf[tid];
}

// ---------------------------------------------------------------------------
extern "C" void kernel_launch(void* const* d_in, const int* in_sizes, int n_in,
                              void* d_out, int out_size, void* d_ws, size_t ws_size,
                              hipStream_t stream)
{
    const float* x = (const float*)d_in[0];   // [256,32,6,6,8]
    const float* W = (const float*)d_in[1];   // [1,32,6,6,10,8,16]
    float* out  = (float*)d_out;              // [256,10,16]
    float* uhat = (float*)d_ws;               // [256,1152,10,16] = 180 MiB

    // 1152 positions * 16 batch-tiles = 18432 waves, 8 waves/block
    uhat_wmma_kernel<<<18432 / 8, 256, 0, stream>>>(x, W, uhat);
    routing_kernel<<<BATCH, 384, 0, stream>>>(uhat, out);
}